// GCNNet_26379689132134
// MI455X (gfx1250) — compile-verified
//
#include <hip/hip_runtime.h>
#include <math.h>

typedef __attribute__((ext_vector_type(2))) float v2f;
typedef __attribute__((ext_vector_type(8))) float v8f;

// ---------------------------------------------------------------------------
// init: zero scatter accumulators, deg = 1.0 (self loop)
// ---------------------------------------------------------------------------
__global__ void gcn_init(float* __restrict__ acc1, float* __restrict__ acc2,
                         float* __restrict__ deg, int n) {
  int gid = blockIdx.x * blockDim.x + threadIdx.x;
  if (gid < n * 64) acc1[gid] = 0.0f;
  if (gid < n * 10) acc2[gid] = 0.0f;
  if (gid < n)      deg[gid]  = 1.0f;
}

// degree accumulation over target nodes (col)
__global__ void gcn_degree(const int* __restrict__ col, float* __restrict__ deg, int E) {
  int e = blockIdx.x * blockDim.x + threadIdx.x;
  if (e < E) atomicAdd(&deg[col[e]], 1.0f);
}

__global__ void gcn_rsqrt(float* __restrict__ deg, int n) {
  int i = blockIdx.x * blockDim.x + threadIdx.x;
  if (i < n) deg[i] = rsqrtf(deg[i]);
}

// ---------------------------------------------------------------------------
// GEMM1: tmp1[N,64] = x[N,128] @ W1[128,64]   (WMMA f32 16x16x4, full fp32)
// One wave -> 16 rows x 64 cols (4 accumulators), K loop of 32 steps.
// A frag (16x4 f32): lane m=L&15 holds row row0+m; half=L>>4 selects K pair.
// B/C/D: row striped across lanes within a VGPR.
// ---------------------------------------------------------------------------
__global__ void gcn_gemm1(const float* __restrict__ x, const float* __restrict__ W1,
                          float* __restrict__ tmp1, int n) {
  int wave = blockIdx.x * (blockDim.x >> 5) + ((int)threadIdx.x >> 5);
  int lane = threadIdx.x & 31;
  int nTiles = (n + 15) >> 4;
  if (wave >= nTiles) return;
  int row0 = wave << 4;
  int half = lane >> 4;      // 0 or 1
  int m    = lane & 15;
  int rA = row0 + m; if (rA > n - 1) rA = n - 1;   // clamp (N%16==0 normally)
  const float* xrow = x + (size_t)rA * 128 + (half << 1);

  v8f c0 = {}, c1 = {}, c2 = {}, c3 = {};
#pragma unroll 4
  for (int k = 0; k < 128; k += 4) {
    v2f a = *(const v2f*)(xrow + k);           // K = k+2h, k+2h+1
    int kb = k + (half << 1);
    const float* w = W1 + (size_t)kb * 64 + m;
    v2f b0 = { w[0],  w[64 + 0]  };
    v2f b1 = { w[16], w[64 + 16] };
    v2f b2 = { w[32], w[64 + 32] };
    v2f b3 = { w[48], w[64 + 48] };
    c0 = __builtin_amdgcn_wmma_f32_16x16x4_f32(false, a, false, b0, (short)0, c0, false, false);
    c1 = __builtin_amdgcn_wmma_f32_16x16x4_f32(false, a, false, b1, (short)0, c1, false, false);
    c2 = __builtin_amdgcn_wmma_f32_16x16x4_f32(false, a, false, b2, (short)0, c2, false, false);
    c3 = __builtin_amdgcn_wmma_f32_16x16x4_f32(false, a, false, b3, (short)0, c3, false, false);
  }
#pragma unroll
  for (int v = 0; v < 8; ++v) {
    int r = row0 + (half << 3) + v;            // M = v + 8*half
    if (r < n) {
      float* o = tmp1 + (size_t)r * 64 + m;
      o[0] = c0[v]; o[16] = c1[v]; o[32] = c2[v]; o[48] = c3[v];
    }
  }
}

// ---------------------------------------------------------------------------
// Layer-1 scatter: acc1[col] += tmp1[row] * dinv[row]*dinv[col]  (64 thr/edge)
// ---------------------------------------------------------------------------
__global__ void gcn_scatter1(const int* __restrict__ row, const int* __restrict__ col,
                             const float* __restrict__ dinv, const float* __restrict__ tmp1,
                             float* __restrict__ acc1, int E) {
  long long gid = (long long)blockIdx.x * blockDim.x + threadIdx.x;
  int e = (int)(gid >> 6);
  int f = (int)(gid & 63);
  if (e >= E) return;
  int r = row[e], c = col[e];
  float w = dinv[r] * dinv[c];
  atomicAdd(&acc1[(size_t)c * 64 + f], tmp1[(size_t)r * 64 + f] * w);
}

// h1 = relu(acc1 + tmp1*dinv^2 (self loop) + b1) -> written into d_out region 2
__global__ void gcn_h1(const float* __restrict__ acc1, const float* __restrict__ tmp1,
                       const float* __restrict__ dinv, const float* __restrict__ b1,
                       float* __restrict__ h1, int n) {
  int gid = blockIdx.x * blockDim.x + threadIdx.x;
  if (gid >= n * 64) return;
  int node = gid >> 6, f = gid & 63;
  float s = dinv[node]; s *= s;
  float v = acc1[gid] + tmp1[gid] * s + b1[f];
  h1[gid] = v > 0.0f ? v : 0.0f;
}

// ---------------------------------------------------------------------------
// GEMM2: tmp2[N,16] = h1[N,64] @ W2pad[64,16]  (cols 10..15 zero-padded)
// ---------------------------------------------------------------------------
__global__ void gcn_gemm2(const float* __restrict__ h1, const float* __restrict__ W2,
                          float* __restrict__ tmp2, int n) {
  int wave = blockIdx.x * (blockDim.x >> 5) + ((int)threadIdx.x >> 5);
  int lane = threadIdx.x & 31;
  int nTiles = (n + 15) >> 4;
  if (wave >= nTiles) return;
  int row0 = wave << 4;
  int half = lane >> 4;
  int m    = lane & 15;
  int rA = row0 + m; if (rA > n - 1) rA = n - 1;
  const float* hrow = h1 + (size_t)rA * 64 + (half << 1);

  v8f c0 = {};
#pragma unroll
  for (int k = 0; k < 64; k += 4) {
    v2f a = *(const v2f*)(hrow + k);
    int kb = k + (half << 1);
    v2f b;
    b.x = (m < 10) ? W2[(size_t)kb * 10 + m]       : 0.0f;  // predicated load OK;
    b.y = (m < 10) ? W2[(size_t)(kb + 1) * 10 + m] : 0.0f;  // only WMMA needs EXEC all-1s
    c0 = __builtin_amdgcn_wmma_f32_16x16x4_f32(false, a, false, b, (short)0, c0, false, false);
  }
#pragma unroll
  for (int v = 0; v < 8; ++v) {
    int r = row0 + (half << 3) + v;
    if (r < n) tmp2[(size_t)r * 16 + m] = c0[v];
  }
}

// Layer-2 scatter: acc2[col,0..9] += tmp2[row]*norm  (16 thr/edge, f<10)
__global__ void gcn_scatter2(const int* __restrict__ row, const int* __restrict__ col,
                             const float* __restrict__ dinv, const float* __restrict__ tmp2,
                             float* __restrict__ acc2, int E) {
  long long gid = (long long)blockIdx.x * blockDim.x + threadIdx.x;
  int e = (int)(gid >> 4);
  int f = (int)(gid & 15);
  if (e >= E || f >= 10) return;
  int r = row[e], c = col[e];
  float w = dinv[r] * dinv[c];
  atomicAdd(&acc2[(size_t)c * 10 + f], tmp2[(size_t)r * 16 + f] * w);
}

// out = log_softmax(acc2 + tmp2*dinv^2 + b2), one thread per node
__global__ void gcn_out(const float* __restrict__ acc2, const float* __restrict__ tmp2,
                        const float* __restrict__ dinv, const float* __restrict__ b2,
                        float* __restrict__ out, int n) {
  int node = blockIdx.x * blockDim.x + threadIdx.x;
  if (node >= n) return;
  float s = dinv[node]; s *= s;
  float v[10];
  float mx = -3.0e38f;
#pragma unroll
  for (int f = 0; f < 10; ++f) {
    v[f] = acc2[(size_t)node * 10 + f] + tmp2[(size_t)node * 16 + f] * s + b2[f];
    mx = fmaxf(mx, v[f]);
  }
  float sum = 0.0f;
#pragma unroll
  for (int f = 0; f < 10; ++f) sum += expf(v[f] - mx);
  float lse = logf(sum);
#pragma unroll
  for (int f = 0; f < 10; ++f) out[(size_t)node * 10 + f] = v[f] - mx - lse;
}

// ---------------------------------------------------------------------------
extern "C" void kernel_launch(void* const* d_in, const int* in_sizes, int n_in,
                              void* d_out, int out_size, void* d_ws, size_t ws_size,
                              hipStream_t stream) {
  const float* x  = (const float*)d_in[0];
  const int*   ei = (const int*)  d_in[1];
  const float* W1 = (const float*)d_in[2];
  const float* b1 = (const float*)d_in[3];
  const float* W2 = (const float*)d_in[4];
  const float* b2 = (const float*)d_in[5];

  int N = in_sizes[0] / 128;
  int E = in_sizes[1] / 2;
  const int* row = ei;        // sources
  const int* col = ei + E;    // targets

  float* out = (float*)d_out;            // [N,10] log-softmax
  float* h1  = out + (size_t)N * 10;     // [N,64] embeddings (output #2)

  float* ws   = (float*)d_ws;
  float* deg  = ws;                      // N      (becomes dinv in place)
  float* tmp1 = deg  + N;                // N*64   x@W1
  float* acc1 = tmp1 + (size_t)N * 64;   // N*64   scatter accumulator
  float* tmp2 = acc1 + (size_t)N * 64;   // N*16   h1@W2 (padded)
  float* acc2 = tmp2 + (size_t)N * 16;   // N*10   scatter accumulator

  const int B = 256;
  int nTiles  = (N + 15) >> 4;                     // 16-row WMMA tiles
  int gemmBlk = (nTiles + (B / 32) - 1) / (B / 32);

  gcn_init  <<<(N * 64 + B - 1) / B, B, 0, stream>>>(acc1, acc2, deg, N);
  gcn_degree<<<(E + B - 1) / B,      B, 0, stream>>>(col, deg, E);
  gcn_rsqrt <<<(N + B - 1) / B,      B, 0, stream>>>(deg, N);

  gcn_gemm1 <<<gemmBlk, B, 0, stream>>>(x, W1, tmp1, N);
  {
    long long t = (long long)E * 64;
    gcn_scatter1<<<(int)((t + B - 1) / B), B, 0, stream>>>(row, col, deg, tmp1, acc1, E);
  }
  gcn_h1    <<<(N * 64 + B - 1) / B, B, 0, stream>>>(acc1, tmp1, deg, b1, h1, N);

  gcn_gemm2 <<<gemmBlk, B, 0, stream>>>(h1, W2, tmp2, N);
  {
    long long t = (long long)E * 16;
    gcn_scatter2<<<(int)((t + B - 1) / B), B, 0, stream>>>(row, col, deg, tmp2, acc2, E);
  }
  gcn_out   <<<(N + B - 1) / B, B, 0, stream>>>(acc2, tmp2, deg, b2, out, N);
}